// gnn_42460046688469
// MI455X (gfx1250) — compile-verified
//
#include <hip/hip_runtime.h>

// Problem sizes (fixed by the reference).
constexpr int Mdim = 1 << 20;   // coefficient rows
constexpr int Kdim = 8;         // neighbors
constexpr int Ndim = 1 << 19;   // GNN nodes
constexpr int Hdim = 1 << 19;   // masked rows
constexpr int Cdim = 16;        // channels
constexpr int Rtot = Ndim + Hdim;

#define ROWS_PER_BLOCK 64
#define THREADS 256

typedef float f4 __attribute__((ext_vector_type(4)));

// LDS layout in dwords (dynamic shared memory, starts at groupstaticsize()):
//   [0,64)      s_idx : hier_ind values for this block's 64 rows
//   [64,576)    ni    : NI[k, s] per (row,k)   (or hier_mask row for copy rows, at k==0)
//   [576,1088)  wre   : w_real[k, s]
//   [1088,1600) wim   : w_imag[k, s]
#define LDS_DWORDS 1600

// CDNA5 async gather into LDS: per-lane global address -> per-lane LDS offset.
// Tracked by ASYNCcnt. Inline asm per CDNA5 ISA (portable
// across ROCm 7.2 / amdgpu-toolchain, bypasses builtin arity differences).
__device__ __forceinline__ void async_gather_b32(const void* gaddr, unsigned lds_byte_off) {
    asm volatile("global_load_async_to_lds_b32 %0, %1, off"
                 :: "v"(lds_byte_off), "v"(gaddr)
                 : "memory");
}

__device__ __forceinline__ void wait_async0() {
#if defined(__has_builtin) && __has_builtin(__builtin_amdgcn_s_wait_asynccnt)
    __builtin_amdgcn_s_wait_asynccnt(0);
#else
    asm volatile("s_wait_asynccnt 0" ::: "memory");
#endif
}

__global__ __launch_bounds__(THREADS) void gnn_fused_kernel(
    const float* __restrict__ Xf_real,   // (M, 1, 16)
    const float* __restrict__ Xf_imag,   // (M, 1, 16)
    const float* __restrict__ w_real,    // (K, N)
    const float* __restrict__ w_imag,    // (K, N)
    const int*   __restrict__ NI,        // (K, N)
    const int*   __restrict__ hier_mask, // (H,)
    const int*   __restrict__ hier_ind,  // (N+H,)
    float*       __restrict__ out)       // (N+H, 16) complex -> interleaved f32
{
    extern __shared__ int lds[];
    const unsigned ldsBase = __builtin_amdgcn_groupstaticsize(); // dynamic LDS byte base
    const int tid = threadIdx.x;
    const int r0  = blockIdx.x * ROWS_PER_BLOCK;

    // ---- Phase A: async-stage this block's 64 permutation entries ----
    if (tid < ROWS_PER_BLOCK) {
        async_gather_b32(hier_ind + r0 + tid, ldsBase + 4u * (unsigned)tid);
    }
    wait_async0();
    __syncthreads();

    // ---- Phase B: async-gather scattered NI / w columns (4B gathers) ----
#pragma unroll
    for (int i = 0; i < 2; ++i) {
        const int p   = tid + i * THREADS;       // (row,k) pair id, 0..511
        const int row = p >> 3;
        const int k   = p & 7;
        const int s   = lds[row];
        if (s < Ndim) {
            const int idx = k * Ndim + s;
            async_gather_b32(NI     + idx, ldsBase + 4u * (unsigned)(64   + p));
            async_gather_b32(w_real + idx, ldsBase + 4u * (unsigned)(576  + p));
            async_gather_b32(w_imag + idx, ldsBase + 4u * (unsigned)(1088 + p));
        } else if (k == 0) {
            async_gather_b32(hier_mask + (s - Ndim), ldsBase + 4u * (unsigned)(64 + p));
        }
    }
    wait_async0();
    __syncthreads();

    // ---- Phase C: complex K-reduction with 128-bit row gathers ----
    const int row = tid >> 2;      // 0..63 : output row within block
    const int q   = tid & 3;       // channel quad: channels 4q..4q+3
    const int s   = lds[row];

    f4 re = {0.f, 0.f, 0.f, 0.f};
    f4 im = {0.f, 0.f, 0.f, 0.f};

    if (s < Ndim) {
        const int*   nrow = lds + 64 + (row << 3);
        const float* wr   = ((const float*)lds) + 576  + (row << 3);
        const float* wi   = ((const float*)lds) + 1088 + (row << 3);
#pragma unroll
        for (int k = 0; k < Kdim; ++k) {
            const int   m = nrow[k];
            const float a = wr[k];
            const float b = wi[k];
            const f4 xr = *(const f4*)(Xf_real + ((size_t)m << 4) + (q << 2));
            const f4 xi = *(const f4*)(Xf_imag + ((size_t)m << 4) + (q << 2));
            re += xr * a - xi * b;   // (xr + i*xi)(a + i*b)
            im += xr * b + xi * a;
        }
    } else {
        const int m = lds[64 + (row << 3)];
        re = *(const f4*)(Xf_real + ((size_t)m << 4) + (q << 2));
        im = *(const f4*)(Xf_imag + ((size_t)m << 4) + (q << 2));
    }

    // Interleave to complex64 layout and stream out (non-temporal: keep Xf in L2).
    const int r = r0 + row;
    f4 o0 = {re.x, im.x, re.y, im.y};
    f4 o1 = {re.z, im.z, re.w, im.w};
    f4* op = (f4*)(out + ((size_t)r << 5) + (q << 3));
    __builtin_nontemporal_store(o0, op);
    __builtin_nontemporal_store(o1, op + 1);
}

extern "C" void kernel_launch(void* const* d_in, const int* in_sizes, int n_in,
                              void* d_out, int out_size, void* d_ws, size_t ws_size,
                              hipStream_t stream) {
    (void)in_sizes; (void)n_in; (void)d_ws; (void)ws_size; (void)out_size;
    const float* Xf_real   = (const float*)d_in[0];
    const float* Xf_imag   = (const float*)d_in[1];
    const float* w_real    = (const float*)d_in[2];
    const float* w_imag    = (const float*)d_in[3];
    const int*   NI        = (const int*)d_in[4];
    const int*   hier_mask = (const int*)d_in[5];
    const int*   hier_ind  = (const int*)d_in[6];
    // d_in[7] = inc (== C == 16), compile-time constant here.
    float* out = (float*)d_out;

    const int grid = Rtot / ROWS_PER_BLOCK;  // 16384 blocks x 256 threads
    gnn_fused_kernel<<<dim3(grid), dim3(THREADS), LDS_DWORDS * 4, stream>>>(
        Xf_real, Xf_imag, w_real, w_imag, NI, hier_mask, hier_ind, out);
}